// CTProjector_ParallelBeam2D_54906861912127
// MI455X (gfx1250) — compile-verified
//
#include <hip/hip_runtime.h>
#include <stdint.h>

// ---------------- problem constants ----------------
#define NXX 512
#define NYY 512
#define NUU 768
#define NTH 180
#define TOTOUT (NTH * NUU)          // 138240 outputs
#define SCHUNK 8                     // row chunks (partial sinograms)
#define ROWS_PER_CHUNK (NYY / SCHUNK)  // 64
#define TILE_R 8                     // rows per LDS tile
#define NTILES (ROWS_PER_CHUNK / TILE_R)
#define LDS_STRIDE 516               // 512 + 4 pad floats (bank-conflict avoidance)

typedef __attribute__((ext_vector_type(2))) float v2f;
typedef __attribute__((ext_vector_type(8))) float v8f;
typedef __attribute__((ext_vector_type(4))) int v4i;

// ---------------- CDNA5 feature probes ----------------
#if defined(__has_builtin)
#if __has_builtin(__builtin_amdgcn_global_load_async_to_lds_b128) && \
    __has_builtin(__builtin_amdgcn_s_wait_asynccnt)
#define HAVE_ASYNC 1
#endif
#if __has_builtin(__builtin_amdgcn_wmma_f32_16x16x4_f32)
#define HAVE_WMMA4 1
#endif
#endif
#ifndef HAVE_ASYNC
#define HAVE_ASYNC 0
#endif
#ifndef HAVE_WMMA4
#define HAVE_WMMA4 0
#endif

// Trapezoid footprint weight of pixel (u0 center) against detector bin center ub.
// Exactly zero outside the footprint support, mirroring the reference formula.
__device__ __forceinline__ float trap_w(float u0, float ub, float b1h, float b2h,
                                        float h, float cA) {
  const float lo = ub - 0.5f, hi = ub + 0.5f;
  const float u1 = u0 - b2h, u2 = u0 - b1h, u3 = u0 + b1h, u4 = u0 + b2h;
  float w = 0.0f;
  {  // rising edge
    float a = fmaxf(u1, lo), b = fminf(u2, hi);
    if (b > a) { float p = b - u1, q = a - u1; w += cA * (p * p - q * q); }
  }
  {  // flat top
    float a = fmaxf(u2, lo), b = fminf(u3, hi);
    if (b > a) w += h * (b - a);
  }
  {  // falling edge
    float a = fmaxf(u3, lo), b = fminf(u4, hi);
    if (b > a) { float p = a - u4, q = b - u4; w += cA * (p * p - q * q); }
  }
  return w;
}

// Forward projector, gather form. grid = (NTH angles, SCHUNK row-chunks),
// block = 256 threads; thread owns 3 detector bins (tid, tid+256, tid+512).
__global__ __launch_bounds__(256) void ct_fp_kernel(const float* __restrict__ img,
                                                    const float* __restrict__ theta,
                                                    float* __restrict__ outbuf,
                                                    int use_atomic) {
  __shared__ __align__(16) float tileF[2][TILE_R * LDS_STRIDE];

  const int t = blockIdx.x;
  const int chunk = blockIdx.y;
  const int tid = threadIdx.x;
  const int rowbase = chunk * ROWS_PER_CHUNK;

  const float th = theta[t];
  const float c = cosf(th), s = sinf(th);
  const float ac = fabsf(c), as_ = fabsf(s);
  const float h = fminf(1.0f / ac, 1.0f / as_);
  const float b1 = fabsf(as_ - ac);
  const float b2 = as_ + ac;
  const float b1h = 0.5f * b1, b2h = 0.5f * b2;
  float den = 0.5f * (b2 - b1);
  if (den == 0.0f) den = 1.0f;                 // matches reference's (u1==u2) term
  const float cA = h / (2.0f * den);
  const bool rowdom = (ac >= as_);
  const float invc = 1.0f / c;
  const float invs = 1.0f / s;
  const float wxi = (0.5f + b2h) / ac + 1e-3f; // x half-width (index units)
  const float wyi = (0.5f + b2h) / as_ + 1e-3f;

  float acc[3] = {0.0f, 0.0f, 0.0f};

  // stage one 8x512 row tile into LDS buffer `buf` (4 x b128 per thread)
  auto stage = [&](int tileIdx, int buf) {
#pragma unroll
    for (int j = 0; j < 4; ++j) {
      int cidx = tid + 256 * j;
      int r = cidx >> 7;          // 128 b128-chunks per row
      int c4 = cidx & 127;
      const float* src =
          img + (size_t)(rowbase + tileIdx * TILE_R + r) * NXX + (c4 << 2);
      float* dst = &tileF[buf][r * LDS_STRIDE + (c4 << 2)];
#if HAVE_ASYNC
      __builtin_amdgcn_global_load_async_to_lds_b128((v4i*)src, (v4i*)dst, 0, 0);
#else
      const float4 v = *(const float4*)src;
      *(float4*)dst = v;
#endif
    }
  };

  stage(0, 0);
  for (int tl = 0; tl < NTILES; ++tl) {
    if (tl + 1 < NTILES) stage(tl + 1, (tl + 1) & 1);
#if HAVE_ASYNC
    if (tl + 1 < NTILES) __builtin_amdgcn_s_wait_asynccnt(4);
    else                 __builtin_amdgcn_s_wait_asynccnt(0);
#endif
    __syncthreads();

    const float* T = &tileF[tl & 1][0];
    const int y0t = rowbase + tl * TILE_R;

    if (rowdom) {
      // near-vertical integration band: loop tile rows, ~4 x-taps each
#pragma unroll
      for (int b = 0; b < 3; ++b) {
        const float ub = (float)(tid + 256 * b) - 383.5f;
        float a0 = 0.0f;
        for (int r = 0; r < TILE_R; ++r) {
          const float yp = (float)(y0t + r) - 255.5f;
          const float ys = yp * s;
          const float xi = (ub - ys) * invc + 255.5f;   // band center (x index)
          const int ix0 = (int)floorf(xi - wxi) + 1;
#pragma unroll
          for (int k = 0; k < 4; ++k) {
            const int ix = ix0 + k;
            if (ix >= 0 && ix < NXX) {
              const float xp = (float)ix - 255.5f;
              const float u0 = xp * c + ys;
              a0 += T[r * LDS_STRIDE + ix] * trap_w(u0, ub, b1h, b2h, h, cA);
            }
          }
        }
        acc[b] += a0;
      }
    } else {
      // near-horizontal band: find x-interval intersecting this row tile,
      // then ~4 y-taps per column
      const float A = -c * invs;  // d(y index)/d(x index), |A| <= 1
#pragma unroll
      for (int b = 0; b < 3; ++b) {
        const float ub = (float)(tid + 256 * b) - 383.5f;
        const float B0 = (ub + 255.5f * c) * invs + 255.5f;  // yi at ix=0
        const float ylo = (float)y0t + wyi - 5.0f;
        const float yhi = (float)y0t + wyi + 9.0f;
        int ixa, ixb;
        if (fabsf(A) < 1e-6f) {
          const bool in = (B0 >= ylo) && (B0 <= yhi);
          ixa = 0;
          ixb = in ? (NXX - 1) : -1;
        } else {
          const float x1 = (ylo - B0) / A;
          const float x2 = (yhi - B0) / A;
          float xa = fminf(x1, x2) - 1.0f;
          float xb = fmaxf(x1, x2) + 1.0f;
          xa = fmaxf(xa, 0.0f);
          xb = fminf(xb, (float)(NXX - 1));
          ixa = (int)xa;
          ixb = (int)xb;
          if (xb < xa) ixb = ixa - 1;  // empty
        }
        float a0 = 0.0f;
        for (int ix = ixa; ix <= ixb; ++ix) {
          const float yi = A * (float)ix + B0;
          const int iy0 = (int)floorf(yi - wyi) + 1;
          const float xp = (float)ix - 255.5f;
          const float xc_ = xp * c;
#pragma unroll
          for (int k = 0; k < 4; ++k) {
            const int iy = iy0 + k;
            const int rr = iy - y0t;
            if (rr >= 0 && rr < TILE_R) {
              const float yp = (float)iy - 255.5f;
              const float u0 = xc_ + yp * s;
              a0 += T[rr * LDS_STRIDE + ix] * trap_w(u0, ub, b1h, b2h, h, cA);
            }
          }
        }
        acc[b] += a0;
      }
    }
    __syncthreads();
  }

#pragma unroll
  for (int b = 0; b < 3; ++b) {
    const int i = tid + 256 * b;
    const size_t o = (size_t)t * NUU + i;
    if (use_atomic) atomicAdd(&outbuf[o], acc[b]);
    else            outbuf[(size_t)chunk * TOTOUT + o] = acc[b];
  }
}

// 8-way partial-sinogram reduction on the matrix pipe:
// D = 1(16x4) x B(4x16) + C sums 4 partials per WMMA; chain two WMMAs for 8.
// All-ones A makes the result independent of the exact B lane/VGPR bijection.
__global__ __launch_bounds__(256) void ct_reduce_kernel(const float* __restrict__ ws,
                                                        float* __restrict__ out) {
  const int gwave = (int)((blockIdx.x * 256u + threadIdx.x) >> 5);
  const int lane = threadIdx.x & 31;
  const int base = gwave << 4;          // 16 outputs per wave
  const int idx = base + (lane & 15);
  const int hi = lane >> 4;             // lanes 16-31 carry the other two rows
#if HAVE_WMMA4
  v2f a; a.x = 1.0f; a.y = 1.0f;
  const int s0 = hi ? 2 : 0;
  v2f bv0, bv1;
  bv0.x = ws[(size_t)(s0 + 0) * TOTOUT + idx];
  bv0.y = ws[(size_t)(s0 + 1) * TOTOUT + idx];
  bv1.x = ws[(size_t)(s0 + 4) * TOTOUT + idx];
  bv1.y = ws[(size_t)(s0 + 5) * TOTOUT + idx];
  v8f cacc = {};
  cacc = __builtin_amdgcn_wmma_f32_16x16x4_f32(false, a, false, bv0, (short)0,
                                               cacc, false, false);
  cacc = __builtin_amdgcn_wmma_f32_16x16x4_f32(false, a, false, bv1, (short)0,
                                               cacc, false, false);
  if (lane < 16) out[idx] = cacc[0];    // D row M=0, col N=lane
#else
  if (lane < 16) {
    float sum = 0.0f;
    for (int s = 0; s < SCHUNK; ++s) sum += ws[(size_t)s * TOTOUT + idx];
    out[idx] = sum;
  }
#endif
}

__global__ void ct_zero_kernel(float* p, int n) {
  int i = blockIdx.x * blockDim.x + threadIdx.x;
  if (i < n) p[i] = 0.0f;
}

extern "C" void kernel_launch(void* const* d_in, const int* in_sizes, int n_in,
                              void* d_out, int out_size, void* d_ws, size_t ws_size,
                              hipStream_t stream) {
  (void)in_sizes; (void)n_in; (void)out_size;
  const float* img = (const float*)d_in[0];     // [512*512] f32
  const float* theta = (const float*)d_in[1];   // [180] f32
  float* out = (float*)d_out;                   // [180*768] f32
  float* ws = (float*)d_ws;

  const size_t need = (size_t)SCHUNK * TOTOUT * sizeof(float);
  if (ws_size >= need) {
    // deterministic path: partial sinograms in scratch + WMMA reduction
    ct_fp_kernel<<<dim3(NTH, SCHUNK), 256, 0, stream>>>(img, theta, ws, 0);
    ct_reduce_kernel<<<TOTOUT / 16 / 8, 256, 0, stream>>>(ws, out);  // 1080 blocks
  } else {
    // fallback: zero-init output, accumulate with global fp32 atomics
    ct_zero_kernel<<<(TOTOUT + 255) / 256, 256, 0, stream>>>(out, TOTOUT);
    ct_fp_kernel<<<dim3(NTH, SCHUNK), 256, 0, stream>>>(img, theta, out, 1);
  }
}